// FisherInformationMetric_12463995093818
// MI455X (gfx1250) — compile-verified
//
#include <hip/hip_runtime.h>
#include <math.h>

typedef float v2f __attribute__((ext_vector_type(2)));
typedef float v4f __attribute__((ext_vector_type(4)));
typedef float v8f __attribute__((ext_vector_type(8)));

#define B_DIM 1024
#define D_DIM 512
#define C_DIM 1000
#define EPS   1e-8f

// ---------------------------------------------------------------------------
// Kernel 1: logits = params @ W   (M=1024, K=512, N=1000 padded to 63 tiles)
// One wave computes one 16x16 tile via V_WMMA_F32_16X16X4_F32.
// Padded B columns are loaded with the index clamped to C_DIM-1 (in-bounds,
// branch-free); they only affect D columns >= 1000, which are never stored.
// ---------------------------------------------------------------------------
__global__ void fim_gemm_logits(const float* __restrict__ A,   // [1024,512]
                                const float* __restrict__ W,   // [512,1000]
                                float* __restrict__ out) {     // [1024,1000]
    const int m0   = blockIdx.x * 16;
    const int n0   = blockIdx.y * 16;
    const int lane = threadIdx.x & 31;
    const int half = lane >> 4;   // 0: K pair {k0,k0+1}, 1: {k0+2,k0+3}
    const int l15  = lane & 15;

    const int  n   = n0 + l15;
    const bool nok = (n < C_DIM);
    const int  nc  = nok ? n : (C_DIM - 1);        // clamped, always in-bounds

    const float* arow = A + (size_t)(m0 + l15) * D_DIM;
    const float* wcol = W + nc;
    v8f acc = {};

#pragma unroll 4
    for (int k0 = 0; k0 < D_DIM; k0 += 4) {
        const int ka = k0 + half * 2;
        v2f afrag = *(const v2f*)(arow + ka);      // A[m, ka], A[m, ka+1]
        v2f bfrag;
        bfrag.x = wcol[(size_t)ka * C_DIM];        // B[ka,   n]
        bfrag.y = wcol[(size_t)(ka + 1) * C_DIM];  // B[ka+1, n]
        acc = __builtin_amdgcn_wmma_f32_16x16x4_f32(
            false, afrag, false, bfrag, (short)0, acc, false, false);
    }

    if (nok) {
        float* orow = out + (size_t)(m0 + half * 8) * C_DIM + n;
#pragma unroll
        for (int r = 0; r < 8; ++r)
            orow[(size_t)r * C_DIM] = acc[r];      // D: VGPR r -> row r / r+8
    }
}

// ---------------------------------------------------------------------------
// Kernel 2: in-place  logits[b,c] -> 1 - C * softmax(logits[b,:])[c]
// One 256-thread block per row.
// ---------------------------------------------------------------------------
__global__ void fim_dlogits(float* __restrict__ logits) {
    const int b = blockIdx.x;
    float* row = logits + (size_t)b * C_DIM;
    __shared__ float red[256];
    const int t = threadIdx.x;

    float m = -INFINITY;
    for (int c = t; c < C_DIM; c += 256) m = fmaxf(m, row[c]);
    red[t] = m; __syncthreads();
    for (int s = 128; s > 0; s >>= 1) {
        if (t < s) red[t] = fmaxf(red[t], red[t + s]);
        __syncthreads();
    }
    m = red[0]; __syncthreads();

    float sum = 0.0f;
    for (int c = t; c < C_DIM; c += 256) sum += __expf(row[c] - m);
    red[t] = sum; __syncthreads();
    for (int s = 128; s > 0; s >>= 1) {
        if (t < s) red[t] += red[t + s];
        __syncthreads();
    }
    const float scale = (float)C_DIM / red[0];

    for (int c = t; c < C_DIM; c += 256)
        row[c] = 1.0f - scale * __expf(row[c] - m);
}

// ---------------------------------------------------------------------------
// Kernel 3: grads = dlogits @ W^T   (M=1024, K=1000, N=512)
// B[k,n] = W[n,k]  ->  per-lane B-fragment loads are contiguous float2s.
// ---------------------------------------------------------------------------
__global__ void fim_gemm_grads(const float* __restrict__ dl,  // [1024,1000]
                               const float* __restrict__ W,   // [512,1000]
                               float* __restrict__ g) {       // [1024,512]
    const int m0   = blockIdx.x * 16;
    const int n0   = blockIdx.y * 16;
    const int lane = threadIdx.x & 31;
    const int half = lane >> 4;
    const int l15  = lane & 15;

    const float* arow = dl + (size_t)(m0 + l15) * C_DIM;
    const float* wrow = W  + (size_t)(n0 + l15) * C_DIM;
    v8f acc = {};

#pragma unroll 4
    for (int k0 = 0; k0 < C_DIM; k0 += 4) {      // 1000 = 250 * 4, exact
        const int ka = k0 + half * 2;
        v2f afrag = *(const v2f*)(arow + ka);    // dl[m, ka..ka+1]
        v2f bfrag = *(const v2f*)(wrow + ka);    // W[n, ka..ka+1] = B[ka..ka+1, n]
        acc = __builtin_amdgcn_wmma_f32_16x16x4_f32(
            false, afrag, false, bfrag, (short)0, acc, false, false);
    }

    float* orow = g + (size_t)(m0 + half * 8) * D_DIM + (n0 + l15);
#pragma unroll
    for (int r = 0; r < 8; ++r)
        orow[(size_t)r * D_DIM] = acc[r];
}

// ---------------------------------------------------------------------------
// Kernel 4: fisher[b,i,j] = g[b,i]*g[b,j] + eps*(i==j)
// 1.07 GB of pure streaming stores -> float4 non-temporal stores.
// Block handles 16 rows of one b; g row cached in LDS.
// ---------------------------------------------------------------------------
__global__ void fim_outer(const float* __restrict__ g,  // [1024,512]
                          float* __restrict__ out) {    // [1024,512,512]
    const int b      = blockIdx.x;
    const int i_base = blockIdx.y * 16;
    __shared__ float gs[D_DIM];

    for (int t = threadIdx.x; t < D_DIM; t += 256)
        gs[t] = g[(size_t)b * D_DIM + t];
    __syncthreads();

    float* obase = out + (size_t)b * D_DIM * D_DIM;

#pragma unroll
    for (int it = 0; it < 8; ++it) {
        const int idx = it * 256 + threadIdx.x;   // 0..2047
        const int r   = idx >> 7;                 // 0..15
        const int c4  = idx & 127;                // 0..127 (float4 column)
        const int i   = i_base + r;
        const int j   = c4 * 4;
        const float gi = gs[i];
        v4f v;
        v.x = gi * gs[j + 0];
        v.y = gi * gs[j + 1];
        v.z = gi * gs[j + 2];
        v.w = gi * gs[j + 3];
        if ((i >> 2) == c4) v[i & 3] += EPS;      // diagonal epsilon
        __builtin_nontemporal_store(v, (v4f*)(obase + (size_t)i * D_DIM + j));
    }
}

// ---------------------------------------------------------------------------
extern "C" void kernel_launch(void* const* d_in, const int* in_sizes, int n_in,
                              void* d_out, int out_size, void* d_ws, size_t ws_size,
                              hipStream_t stream) {
    const float* params = (const float*)d_in[0];   // [1024,512]
    const float* W      = (const float*)d_in[1];   // [512,1000]
    float* out    = (float*)d_out;                 // [1024,512,512]
    float* logits = (float*)d_ws;                  // 1024*1000 f32 (4.1 MB)
    float* grads  = logits + (size_t)B_DIM * C_DIM;// 1024*512  f32 (2.1 MB)

    // GEMM1: 64 M-tiles x 63 N-tiles (N padded/masked past 1000), 1 wave/tile
    fim_gemm_logits<<<dim3(B_DIM / 16, (C_DIM + 15) / 16), 32, 0, stream>>>(
        params, W, logits);

    // softmax -> dlogits, in place
    fim_dlogits<<<B_DIM, 256, 0, stream>>>(logits);

    // GEMM2: 64 x 32 tiles
    fim_gemm_grads<<<dim3(B_DIM / 16, D_DIM / 16), 32, 0, stream>>>(
        logits, W, grads);

    // Outer products: 1024 x 32 blocks, each streams 32 KB of stores
    fim_outer<<<dim3(B_DIM, D_DIM / 16), 256, 0, stream>>>(grads, out);
}